// HungarianMatcher_91225105367405
// MI455X (gfx1250) — compile-verified
//
#include <hip/hip_runtime.h>
#include <stdint.h>

// Problem constants from the reference: bs=16, nq=900, nc=91, T=1600
#define NC   91
#define NCP  92            // padded class-table stride: 92*4 = 368 bytes = 16B-aligned rows
#define TT   160           // threads per block = target columns per block (5 wave32 waves)
#define TN   32            // prediction rows per block

// 16-byte int vector, matching the async-LDS builtin's pointee type exactly
typedef int v4i __attribute__((vector_size(16)));

// ---------- address-space helpers (generic -> AS1/AS3 via integer round-trip) ----------
__device__ __forceinline__ __attribute__((address_space(1))) v4i* to_as1_v4(const void* p) {
  return (__attribute__((address_space(1))) v4i*)(uintptr_t)p;
}
__device__ __forceinline__ __attribute__((address_space(3))) v4i* to_as3_v4(void* p) {
  // generic LDS address: low 32 bits are the LDS byte offset (flat aperture scheme)
  return (__attribute__((address_space(3))) v4i*)(uint32_t)(uintptr_t)p;
}

#if defined(__has_builtin)
#if __has_builtin(__builtin_amdgcn_global_load_async_to_lds_b128) && \
    __has_builtin(__builtin_amdgcn_s_wait_asynccnt)
#define USE_ASYNC_LDS 1
#endif
#endif

// ---------------- kernel 1: focal class-cost table [N, NCP] ----------------
// table[n,c] = 0.25*(1-p)^2*(-log(p+eps)) - 0.75*p^2*(-log(1-p+eps)), p = sigmoid(logit)
__global__ __launch_bounds__(256) void cls_kernel(const float* __restrict__ logits,
                                                  float* __restrict__ cls, int total) {
  int i = blockIdx.x * 256 + threadIdx.x;
  if (i >= total) return;
  int n  = i / NC;
  int ci = i - n * NC;
  float l   = logits[i];
  float p   = __builtin_amdgcn_rcpf(1.0f + __expf(-l));   // sigmoid
  float omp = 1.0f - p;
  float pos = 0.25f * omp * omp * (-__logf(p   + 1e-8f));
  float neg = 0.75f * p   * p   * (-__logf(omp + 1e-8f));
  cls[(size_t)n * NCP + ci] = pos - neg;
}

// ---------------- kernel 2: per-row box precompute [cx,cy,w,h,x0,y0,x1,y1] ----------------
__global__ __launch_bounds__(256) void box_kernel(const float* __restrict__ pb,
                                                  const float* __restrict__ tb,
                                                  float* __restrict__ p8,
                                                  float* __restrict__ t8, int N, int T) {
  int i = blockIdx.x * 256 + threadIdx.x;
  const float* src;
  float* dst;
  if (i < N)          { src = pb + (size_t)i * 4;       dst = p8 + (size_t)i * 8; }
  else if (i < N + T) { int j = i - N; src = tb + (size_t)j * 4; dst = t8 + (size_t)j * 8; }
  else return;
  float4 b  = *(const float4*)src;
  float  hw = 0.5f * b.z, hh = 0.5f * b.w;
  *(float4*)(dst)     = b;                                              // cx cy w h
  *(float4*)(dst + 4) = make_float4(b.x - hw, b.y - hh, b.x + hw, b.y + hh); // x0 y0 x1 y1
}

// ---------------- kernel 3: pairwise cost matrix [N, T] ----------------
__global__ __launch_bounds__(TT) void cost_kernel(const float* __restrict__ pred8,
                                                  const float* __restrict__ tgt8,
                                                  const int*   __restrict__ tgt_ids,
                                                  const float* __restrict__ cls,
                                                  float* __restrict__ out, int N, int T) {
  __shared__ __align__(16) float scls[TN * NCP];   // 11776 B class-cost tile

  const int tx   = threadIdx.x;
  const int n0   = blockIdx.y * TN;
  const int col  = blockIdx.x * TT + tx;
  const bool colv = (col < T);
  const int  c    = colv ? col : 0;

  // ---- stage class-cost tile (TN x NCP floats, rows 16B-aligned) into LDS ----
  const float* gsrc = cls + (size_t)n0 * NCP;
#ifdef USE_ASYNC_LDS
  for (int i = tx; i < (TN * NCP) / 4; i += TT) {
    __builtin_amdgcn_global_load_async_to_lds_b128(to_as1_v4(gsrc + i * 4),
                                                   to_as3_v4(scls + i * 4),
                                                   /*offset=*/0, /*cpol=*/0);
  }
  __builtin_amdgcn_s_wait_asynccnt(0);   // this wave's async copies landed in LDS
#else
  for (int i = tx; i < (TN * NCP) / 4; i += TT) {
    const float4 v = *(const float4*)(gsrc + i * 4);
    *(float4*)(scls + i * 4) = v;
  }
#endif

  // ---- per-column target data held in registers for the whole row loop ----
  const float4 tA = *(const float4*)(tgt8 + (size_t)c * 8);      // cx cy w h
  const float4 tB = *(const float4*)(tgt8 + (size_t)c * 8 + 4);  // x0 y0 x1 y1
  const float  tarea = tA.z * tA.w;
  const int    tid   = tgt_ids[c];

  __syncthreads();   // all waves' tiles visible

  const int rmax = (N - n0 < TN) ? (N - n0) : TN;
#pragma unroll 4
  for (int r = 0; r < rmax; ++r) {
    const int n = n0 + r;
    const float4 pA = *(const float4*)(pred8 + (size_t)n * 8);      // uniform across block
    const float4 pB = *(const float4*)(pred8 + (size_t)n * 8 + 4);
    const float parea = pA.z * pA.w;

    // L1 in cxcywh space
    float l1 = fabsf(pA.x - tA.x) + fabsf(pA.y - tA.y)
             + fabsf(pA.z - tA.z) + fabsf(pA.w - tA.w);

    // GIoU
    float iw = fminf(pB.z, tB.z) - fmaxf(pB.x, tB.x);
    float ih = fminf(pB.w, tB.w) - fmaxf(pB.y, tB.y);
    iw = fmaxf(iw, 0.0f); ih = fmaxf(ih, 0.0f);
    const float inter = iw * ih;
    const float uni   = parea + tarea - inter;
    const float iou   = inter * __builtin_amdgcn_rcpf(uni);

    float ew = fmaxf(pB.z, tB.z) - fminf(pB.x, tB.x);
    float eh = fmaxf(pB.w, tB.w) - fminf(pB.y, tB.y);
    ew = fmaxf(ew, 0.0f); eh = fmaxf(eh, 0.0f);
    const float earea = ew * eh;
    const float giou  = iou - (earea - uni) * __builtin_amdgcn_rcpf(earea);

    const float cc = scls[r * NCP + tid];   // ds_load_b32 gather from staged tile

    if (colv) out[(size_t)n * T + col] = 5.0f * l1 + 2.0f * cc - 2.0f * giou;
  }
}

extern "C" void kernel_launch(void* const* d_in, const int* in_sizes, int n_in,
                              void* d_out, int out_size, void* d_ws, size_t ws_size,
                              hipStream_t stream) {
  const float* logits = (const float*)d_in[0];   // [bs, nq, nc] f32
  const float* pbox   = (const float*)d_in[1];   // [bs, nq, 4]  f32
  const int*   tids   = (const int*)d_in[2];     // [T] int32
  const float* tbox   = (const float*)d_in[3];   // [T, 4] f32
  float* out = (float*)d_out;                    // [bs, nq, T] f32

  const int N = in_sizes[1] / 4;   // 14400
  const int T = in_sizes[3] / 4;   // 1600

  // workspace layout
  char*  ws       = (char*)d_ws;
  size_t clsBytes = (((size_t)N * NCP * sizeof(float)) + 255) & ~(size_t)255;
  size_t p8Bytes  = (((size_t)N * 8 * sizeof(float)) + 255) & ~(size_t)255;
  float* cls = (float*)ws;
  float* p8  = (float*)(ws + clsBytes);
  float* t8  = (float*)(ws + clsBytes + p8Bytes);

  const int total = N * NC;
  cls_kernel<<<(total + 255) / 256, 256, 0, stream>>>(logits, cls, total);
  box_kernel<<<(N + T + 255) / 256, 256, 0, stream>>>(pbox, tbox, p8, t8, N, T);

  dim3 grid((T + TT - 1) / TT, (N + TN - 1) / TN);
  cost_kernel<<<grid, TT, 0, stream>>>(p8, t8, tids, cls, out, N, T);
}